// RGCNLayer_84662395339200
// MI455X (gfx1250) — compile-verified
//
#include <hip/hip_runtime.h>
#include <hip/hip_bf16.h>

#define N_NODES 20000
#define N_EDGES 320000
#define NUM_RELS 8
#define DIM 256
#define BN_EPS 1e-5f
#define ROW_TILES (N_NODES / 16)   // 1250

typedef __attribute__((ext_vector_type(2))) float v2f;
typedef __attribute__((ext_vector_type(8))) float v8f;

// ---------------------------------------------------------------------------
// Scatter: S[dst] += h[src] for edges of relation r. One wave per edge,
// each lane handles 8 features (stride-32 => coalesced 128B runs).
// ---------------------------------------------------------------------------
__global__ void rgcn_scatter(const float* __restrict__ h,
                             const int* __restrict__ src,
                             const int* __restrict__ dst,
                             const int* __restrict__ rel,
                             float* __restrict__ S, int r) {
    const int lane = threadIdx.x & 31;
    const int e = blockIdx.x * 8 + (threadIdx.x >> 5);
    if (e >= N_EDGES) return;
    if (rel[e] != r) return;
    const float* __restrict__ hs = h + (size_t)src[e] * DIM;
    float* __restrict__ sd = S + (size_t)dst[e] * DIM;
#pragma unroll
    for (int j = 0; j < 8; ++j) {
        atomicAdd(&sd[lane + 32 * j], hs[lane + 32 * j]);
    }
}

// ---------------------------------------------------------------------------
// agg[M,256] += S[M,256] @ W[256,256]   (row-major, W = weight[r])
// One wave computes a 16x64 output strip (4 WMMA tiles), K-loop step 4 using
// V_WMMA_F32_16X16X4_F32. A fragment reused across the 4 column tiles.
// A layout (ISA 7.12.2, 32-bit 16x4): lanes 0-15 hold K=0/1, lanes 16-31 K=2/3.
// C/D layout: vgpr v -> M = v (lanes 0-15) or v+8 (lanes 16-31), N = lane&15.
// ---------------------------------------------------------------------------
__global__ void rgcn_gemm_acc(const float* __restrict__ S,
                              const float* __restrict__ W,
                              float* __restrict__ agg) {
    const int lane = threadIdx.x & 31;
    const int wv = threadIdx.x >> 5;
    const int rowTile = blockIdx.y * 8 + wv;
    if (rowTile >= ROW_TILES) return;               // whole-wave exit: EXEC stays all-1s
    const int row0 = rowTile * 16;
    const int col0 = blockIdx.x * 64;
    const int mn = lane & 15;                       // M index for A / N index for B
    const int kh = (lane >> 4) << 1;                // 0 or 2: which K pair this half-wave holds
    const int cm = (lane >> 4) << 3;                // 0 or 8: C/D row offset for upper half

    // beta = 1: preload current accumulator tiles
    v8f c0, c1, c2, c3;
#pragma unroll
    for (int v = 0; v < 8; ++v) {
        const float* p = agg + (size_t)(row0 + cm + v) * DIM + col0 + mn;
        c0[v] = p[0];
        c1[v] = p[16];
        c2[v] = p[32];
        c3[v] = p[48];
    }

    const float* __restrict__ Sr = S + (size_t)(row0 + mn) * DIM;
#pragma unroll 4
    for (int k0 = 0; k0 < DIM; k0 += 4) {
        v2f a;
        a.x = Sr[k0 + kh];
        a.y = Sr[k0 + kh + 1];
        const float* __restrict__ Wk0 = W + (size_t)(k0 + kh) * DIM + col0 + mn;
        const float* __restrict__ Wk1 = Wk0 + DIM;
        v2f b0, b1, b2, b3;
        b0.x = Wk0[0];  b0.y = Wk1[0];
        b1.x = Wk0[16]; b1.y = Wk1[16];
        b2.x = Wk0[32]; b2.y = Wk1[32];
        b3.x = Wk0[48]; b3.y = Wk1[48];
        c0 = __builtin_amdgcn_wmma_f32_16x16x4_f32(false, a, false, b0, (short)0, c0, false, false);
        c1 = __builtin_amdgcn_wmma_f32_16x16x4_f32(false, a, false, b1, (short)0, c1, false, false);
        c2 = __builtin_amdgcn_wmma_f32_16x16x4_f32(false, a, false, b2, (short)0, c2, false, false);
        c3 = __builtin_amdgcn_wmma_f32_16x16x4_f32(false, a, false, b3, (short)0, c3, false, false);
    }

#pragma unroll
    for (int v = 0; v < 8; ++v) {
        float* p = agg + (size_t)(row0 + cm + v) * DIM + col0 + mn;
        p[0]  = c0[v];
        p[16] = c1[v];
        p[32] = c2[v];
        p[48] = c3[v];
    }
}

// ---------------------------------------------------------------------------
// Per-column sum / sum-of-squares over 20000 rows. Thread c owns column c,
// blocks stride over rows; one atomicAdd pair per thread at the end.
// ---------------------------------------------------------------------------
__global__ void rgcn_colstats(const float* __restrict__ agg,
                              float* __restrict__ csum,
                              float* __restrict__ csumsq) {
    const int c = threadIdx.x;
    float s = 0.f, s2 = 0.f;
    for (int n = blockIdx.x; n < N_NODES; n += gridDim.x) {
        float x = agg[(size_t)n * DIM + c];
        s += x;
        s2 += x * x;
    }
    atomicAdd(&csum[c], s);
    atomicAdd(&csumsq[c], s2);
}

// ---------------------------------------------------------------------------
// BatchNorm finalize: biased variance = E[x^2] - mean^2, affine.
// ---------------------------------------------------------------------------
__global__ void rgcn_bn(const float* __restrict__ agg,
                        const float* __restrict__ csum,
                        const float* __restrict__ csumsq,
                        const float* __restrict__ gamma,
                        const float* __restrict__ beta,
                        float* __restrict__ out) {
    const int c = threadIdx.x;
    const int n = blockIdx.x;
    const float invN = 1.0f / (float)N_NODES;
    float mean = csum[c] * invN;
    float var = csumsq[c] * invN - mean * mean;
    float inv = rsqrtf(var + BN_EPS);
    float x = agg[(size_t)n * DIM + c];
    out[(size_t)n * DIM + c] = (x - mean) * inv * gamma[c] + beta[c];
}

extern "C" void kernel_launch(void* const* d_in, const int* in_sizes, int n_in,
                              void* d_out, int out_size, void* d_ws, size_t ws_size,
                              hipStream_t stream) {
    const float* h      = (const float*)d_in[0];
    const float* weight = (const float*)d_in[1];
    const float* gamma  = (const float*)d_in[2];
    const float* beta   = (const float*)d_in[3];
    const int*   src    = (const int*)d_in[4];
    const int*   dst    = (const int*)d_in[5];
    const int*   rel    = (const int*)d_in[6];
    float* out = (float*)d_out;

    const size_t matBytes = (size_t)N_NODES * DIM * sizeof(float); // 20.48 MB
    char* ws = (char*)d_ws;
    float* S        = (float*)ws;                      // [N, D] per-relation scatter buffer
    float* agg      = (float*)(ws + matBytes);         // [N, D] accumulated output
    float* csum     = (float*)(ws + 2 * matBytes);     // [D]
    float* csumsq   = csum + DIM;                      // [D]

    // zero agg + stats (contiguous region)
    hipMemsetAsync(agg, 0, matBytes + 2 * DIM * sizeof(float), stream);

    for (int r = 0; r < NUM_RELS; ++r) {
        hipMemsetAsync(S, 0, matBytes, stream);
        rgcn_scatter<<<N_EDGES / 8, 256, 0, stream>>>(h, src, dst, rel, S, r);
        dim3 g(DIM / 64, (ROW_TILES + 7) / 8);
        rgcn_gemm_acc<<<g, 256, 0, stream>>>(S, weight + (size_t)r * DIM * DIM, agg);
    }

    rgcn_colstats<<<128, 256, 0, stream>>>(agg, csum, csumsq);
    rgcn_bn<<<N_NODES, 256, 0, stream>>>(agg, csum, csumsq, gamma, beta, out);
}